// MAHCL_36593121362249
// MI455X (gfx1250) — compile-verified
//
#include <hip/hip_runtime.h>

typedef __attribute__((ext_vector_type(2))) float v2f;
typedef __attribute__((ext_vector_type(4))) float v4f;
typedef __attribute__((ext_vector_type(8))) float v8f;

#define D_FEAT 64

// ---------------- degree / normalization ----------------

__global__ void k_zero_f32(float* p, int n) {
    int i = blockIdx.x * blockDim.x + threadIdx.x;
    if (i < n) p[i] = 0.0f;
}

__global__ void k_deg(const int* __restrict__ eu, const int* __restrict__ ei,
                      float* __restrict__ deg, int E, int n_users) {
    int e = blockIdx.x * blockDim.x + threadIdx.x;
    if (e < E) {
        atomicAdd(&deg[eu[e]], 1.0f);
        atomicAdd(&deg[n_users + ei[e]], 1.0f);
    }
}

__global__ void k_dinv(float* __restrict__ dinv, int n) {
    int i = blockIdx.x * blockDim.x + threadIdx.x;
    if (i < n) dinv[i] = rsqrtf(dinv[i] + 1.0f);  // +1 = self loop
}

// ---------------- init: h = out = concat(user_w, item_w) ----------------

__global__ void k_init(const float* __restrict__ uw, const float* __restrict__ iw,
                       float* __restrict__ h, float* __restrict__ out,
                       long long ub, long long total) {
    long long i = (long long)blockIdx.x * blockDim.x + threadIdx.x;
    if (i < total) {
        float v = (i < ub) ? uw[i] : iw[i - ub];
        h[i] = v;
        out[i] = v;
    }
}

// ---------------- self-loop scaling via WMMA (+ fused acc += h) ----------------
// h_new[v, :] = dinv[v]^2 * h[v, :]  computed as D = diag(s) * H with
// 4 chained V_WMMA_F32_16X16X4_F32 per 16x16 tile.
// One wave per (16-node x 16-feature) tile. EXEC stays all-ones per wave.

__global__ void __launch_bounds__(256)
k_scale_wmma(const float* __restrict__ h, const float* __restrict__ dinv,
             float* __restrict__ hn, float* __restrict__ out,
             int accum, int nrows) {
    int wave = (blockIdx.x * blockDim.x + threadIdx.x) >> 5;
    int lane = threadIdx.x & 31;
    int nodeTiles = (nrows + 15) / 16;
    if (wave >= nodeTiles * 4) return;            // wave-uniform

    int R = (wave >> 2) * 16;                     // node-tile base row
    int C = (wave & 3) * 16;                      // feature-tile base col
    int m  = lane & 15;
    int hi = lane >> 4;                           // lane half: 0 or 1

    if (R + 16 <= nrows) {
        float s = dinv[R + m];
        s = s * s;
        v8f c = {};
#pragma unroll
        for (int g = 0; g < 4; ++g) {
            int k0 = 4 * g + 2 * hi;              // K index of VGPR0 slot for this lane half
            v2f a;                                 // A[m,k] = s[m] iff m == 4g+k
            a.x = (m == k0)     ? s : 0.0f;
            a.y = (m == k0 + 1) ? s : 0.0f;
            const float* p = h + (size_t)(R + k0) * D_FEAT + C + m;
            v2f b;                                 // B[k,j] = H[R+4g+k, C+j]
            b.x = p[0];
            b.y = p[D_FEAT];
            c = __builtin_amdgcn_wmma_f32_16x16x4_f32(
                    false, a, false, b, (short)0, c, false, false);
        }
        // D layout: VGPR v -> row R + v + 8*hi, col C + m
        float* dst = hn + (size_t)(R + 8 * hi) * D_FEAT + C + m;
#pragma unroll
        for (int v = 0; v < 8; ++v) dst[(size_t)v * D_FEAT] = c[v];

        if (accum) {                               // out += h, coalesced remap of tile
            int row = R + (lane >> 1);
            int col = C + (lane & 1) * 8;
            size_t base = (size_t)row * D_FEAT + col;
            const v4f* hp = (const v4f*)(h + base);
            v4f* op = (v4f*)(out + base);
            op[0] = op[0] + hp[0];
            op[1] = op[1] + hp[1];
        }
    } else {
        // partial node tile (not hit for n = 700000): scalar fallback
        for (int r = R + hi; r < nrows; r += 2) {
            float s = dinv[r];
            size_t ix = (size_t)r * D_FEAT + C + m;
            hn[ix] = s * s * h[ix];
            if (accum) out[ix] += h[ix];
        }
    }
}

// ---------------- per-edge propagation (both directions) ----------------
// One wave per edge; lane handles 2 of the 64 features each direction.

__global__ void __launch_bounds__(256)
k_edges(const int* __restrict__ eu, const int* __restrict__ ei,
        const float* __restrict__ dinv, const float* __restrict__ h,
        float* __restrict__ hn, int E, int n_users) {
    int wave = (blockIdx.x * blockDim.x + threadIdx.x) >> 5;
    int lane = threadIdx.x & 31;
    if (wave >= E) return;
    int u  = eu[wave];
    int it = n_users + ei[wave];
    float w = dinv[u] * dinv[it];
    const float2 xu = *(const float2*)(h + (size_t)u  * D_FEAT + lane * 2);
    const float2 xi = *(const float2*)(h + (size_t)it * D_FEAT + lane * 2);
    float* du = hn + (size_t)u  * D_FEAT + lane * 2;
    float* di = hn + (size_t)it * D_FEAT + lane * 2;
    atomicAdd(di,     w * xu.x);
    atomicAdd(di + 1, w * xu.y);
    atomicAdd(du,     w * xi.x);
    atomicAdd(du + 1, w * xi.y);
}

// ---------------- final: interest = (acc + h3)/4; softmax mix for users ----------------

__global__ void k_final(const float* __restrict__ hlast, const float* __restrict__ uw,
                        const float* __restrict__ aspect, float* __restrict__ out,
                        long long ub, long long total) {
    long long i = (long long)blockIdx.x * blockDim.x + threadIdx.x;
    if (i >= total) return;
    float e0 = __expf(aspect[0]);
    float e1 = __expf(aspect[1]);
    float inv = 1.0f / (e0 + e1);
    float a0 = e0 * inv, a1 = e1 * inv;
    float interest = (out[i] + hlast[i]) * 0.25f;
    out[i] = (i < ub) ? (a0 * interest + a1 * uw[i]) : interest;
}

// ---------------- launch ----------------

extern "C" void kernel_launch(void* const* d_in, const int* in_sizes, int n_in,
                              void* d_out, int out_size, void* d_ws, size_t ws_size,
                              hipStream_t stream) {
    const int*   edges  = (const int*)d_in[0];    // (2, E) stacked: users then items
    const float* user_w = (const float*)d_in[1];
    const float* item_w = (const float*)d_in[2];
    const float* aspect = (const float*)d_in[3];

    int E       = in_sizes[0] / 2;
    int n_users = in_sizes[1] / D_FEAT;
    int n_items = in_sizes[2] / D_FEAT;
    int n       = n_users + n_items;
    long long total = (long long)n * D_FEAT;
    long long ub    = (long long)n_users * D_FEAT;

    const int* eu = edges;
    const int* ei = edges + E;

    float* out  = (float*)d_out;
    float* h_a  = (float*)d_ws;                   // n*64 floats
    float* h_b  = h_a + total;                    // n*64 floats
    float* dinv = h_b + total;                    // n floats (also used as deg)

    int eb  = (E + 255) / 256;
    int nb  = (n + 255) / 256;
    int tb  = (int)((total + 255) / 256);
    int edgeBlocks = (E + 7) / 8;                 // 8 waves per 256-thread block
    int tileBlocks = (((n + 15) / 16) * 4 + 7) / 8;

    // normalization: deg (with self loop) -> dinv
    k_zero_f32<<<nb, 256, 0, stream>>>(dinv, n);
    k_deg    <<<eb, 256, 0, stream>>>(eu, ei, dinv, E, n_users);
    k_dinv   <<<nb, 256, 0, stream>>>(dinv, n);

    // h_a = out = x
    k_init<<<tb, 256, 0, stream>>>(user_w, item_w, h_a, out, ub, total);

    // layer 1: h_b = Â h_a          (acc already holds x)
    k_scale_wmma<<<tileBlocks, 256, 0, stream>>>(h_a, dinv, h_b, out, 0, n);
    k_edges     <<<edgeBlocks, 256, 0, stream>>>(eu, ei, dinv, h_a, h_b, E, n_users);
    // layer 2: acc += h1; h_a = Â h_b
    k_scale_wmma<<<tileBlocks, 256, 0, stream>>>(h_b, dinv, h_a, out, 1, n);
    k_edges     <<<edgeBlocks, 256, 0, stream>>>(eu, ei, dinv, h_b, h_a, E, n_users);
    // layer 3: acc += h2; h_b = Â h_a
    k_scale_wmma<<<tileBlocks, 256, 0, stream>>>(h_a, dinv, h_b, out, 1, n);
    k_edges     <<<edgeBlocks, 256, 0, stream>>>(eu, ei, dinv, h_a, h_b, E, n_users);

    // final mix (acc += h3 folded in)
    k_final<<<tb, 256, 0, stream>>>(h_b, user_w, aspect, out, ub, total);
}